// MLPA_LOSS_57105885168195
// MI455X (gfx1250) — compile-verified
//
#include <hip/hip_runtime.h>

// single-instruction hardware transcendentals (base-2): v_log_f32 / v_exp_f32
#define LOG2F(x) __builtin_amdgcn_logf(x)
#define EXP2F(x) __builtin_amdgcn_exp2f(x)

// ---------------- compile-time constants (derived in double) ----------------
constexpr double dM00=0.65, dM01=0.70, dM02=0.29;
constexpr double dM10=0.07, dM11=0.99, dM12=0.11;
constexpr double dM20=0.27, dM21=0.57, dM22=0.78;
constexpr double dDET = dM00*(dM11*dM22 - dM12*dM21)
                      - dM01*(dM10*dM22 - dM12*dM20)
                      + dM02*(dM10*dM21 - dM11*dM20);
// column 2 of inv(RGB_FROM_HED): inv[i][2] = cof(2,i)/det
constexpr double dH2X =  (dM01*dM12 - dM02*dM11) / dDET;
constexpr double dH2Y = -(dM00*dM12 - dM02*dM10) / dDET;
constexpr double dH2Z =  (dM00*dM11 - dM01*dM10) / dDET;

constexpr double dLN10   = 2.302585092994045684;
constexpr double dLN2    = 0.693147180559945309;
constexpr double dLOG2E  = 1.442695040888963407;
constexpr double dLOG102 = 0.301029995663981195;
constexpr double dLOG_ADJUST = -6.0 * dLN10;            // ln(1e-6)

// d = max(K0*log2(r) + K1*log2(g) + K2*log2(b), 0)   (folds /LOG_ADJUST and ln2)
constexpr float K0 = (float)(dH2X * dLN2 / dLOG_ADJUST);
constexpr float K1 = (float)(dH2Y * dLN2 / dLOG_ADJUST);
constexpr float K2 = (float)(dH2Z * dLN2 / dLOG_ADJUST);
// rgb_d[c] = exp2(d * Ec)  (exponent <= 0, so clip[0,1] is automatic)
constexpr float E0 = (float)(dLOG_ADJUST * dM20 * dLOG2E);
constexpr float E1 = (float)(dLOG_ADJUST * dM21 * dLOG2E);
constexpr float E2 = (float)(dLOG_ADJUST * dM22 * dLOG2E);
constexpr float L102    = (float)dLOG102;
constexpr float ADJ_CAL = 0.018075529f;                  // 10^(-e^(1/1.8))
constexpr float INV_BW  = (float)(20.0 / 2.7182818284590452354); // bins / e

// workspace layout (floats): per-wave block partials + per-WG histograms
//   PARTI [16][32][4]  @ 0       (batch, chunk, wave)   in-image
//   PARTT [16][32][4]  @ 2048                           tg-image
//   HISTI [16][32][20] @ 4096    (batch, chunk, bin)
//   HISTT [16][32][20] @ 14336
// total 24576 floats = 96 KB, fully overwritten every launch.
#define WS_PARTI 0
#define WS_PARTT 2048
#define WS_HISTI 4096
#define WS_HISTT 14336

#define HW  262144          // 512*512
#define BATCH 16

__device__ __forceinline__ float fod_of(float r, float g, float b) {
    r = fmaxf(r, 1e-6f); g = fmaxf(g, 1e-6f); b = fmaxf(b, 1e-6f);
    float d = fmaxf(K0 * LOG2F(r) + K1 * LOG2F(g) + K2 * LOG2F(b), 0.0f);
    float rd = EXP2F(d * E0);
    float gd = EXP2F(d * E1);
    float bd = EXP2F(d * E2);
    float grey = fminf(0.2125f * rd + 0.7154f * gd + 0.0721f * bd, 1.0f);
    float v = fmaxf(-L102 * LOG2F(grey + ADJ_CAL), 0.0f);     // log10(1/(grey+adj))
    return (v > 0.0f) ? EXP2F(1.8f * LOG2F(v)) : 0.0f;        // v^1.8
}

// 512 workgroups x 128 threads. WG = (batch b, 16-row chunk). Thread t owns
// pixels w = 4t..4t+3 of each row -> its column-block == wave id, and the 16
// rows stay inside one block-row, so each wave accumulates exactly one block.
__global__ void __launch_bounds__(128)
mlpa_main(const float* __restrict__ inp, const float* __restrict__ tgt,
          float* __restrict__ out, float* __restrict__ ws) {
    __shared__ float histP[2 * 128 * 20];   // per-thread private histograms (20 KB)

    const int t     = threadIdx.x;          // 0..127
    const int wg    = blockIdx.x;           // 0..511
    const int b     = wg >> 5;              // batch
    const int chunk = wg & 31;              // 16-row chunk within image
    const int h0    = chunk << 4;
    const int w0    = t << 2;
    const int wave  = t >> 5;               // == column block 0..3

    #pragma unroll
    for (int img = 0; img < 2; ++img)
        #pragma unroll
        for (int j = 0; j < 20; ++j)
            histP[(img * 128 + t) * 20 + j] = 0.0f;   // private: no barrier needed yet

    const float* srcs[2]  = { inp, tgt };
    float*       mouts[2] = { out + 1 + (size_t)b * HW,
                              out + 1 + (size_t)BATCH * HW + (size_t)b * HW };
    float sums[2] = { 0.0f, 0.0f };

    for (int r = 0; r < 16; ++r) {
        const int off = (h0 + r) * 512 + w0;
        #pragma unroll
        for (int img = 0; img < 2; ++img) {
            const float* base = srcs[img] + (size_t)(3 * b) * HW;
            float4 cr = *(const float4*)(base + off);            // R plane
            float4 cg = *(const float4*)(base + HW + off);       // G plane
            float4 cb = *(const float4*)(base + 2 * HW + off);   // B plane
            const float ra[4] = { cr.x, cr.y, cr.z, cr.w };
            const float ga[4] = { cg.x, cg.y, cg.z, cg.w };
            const float ba[4] = { cb.x, cb.y, cb.z, cb.w };
            #pragma unroll
            for (int j = 0; j < 4; ++j) {
                float f = fod_of(ra[j], ga[j], ba[j]);
                mouts[img][off + j] = (f >= 0.68f) ? 1.0f : 0.0f;   // dword store (+1 offset)
                sums[img] += (f < 0.15f) ? 0.0f : f;                // fod_relu
                if (f > 0.0f) {                                     // weighted histogram
                    int idx = (int)(f * INV_BW);
                    idx = idx > 19 ? 19 : idx;
                    histP[(img * 128 + t) * 20 + idx] += f;
                }
            }
        }
    }

    // wave32 reduction of this wave's block partial (deterministic tree)
    float sIn = sums[0], sTg = sums[1];
    #pragma unroll
    for (int m = 16; m >= 1; m >>= 1) {
        sIn += __shfl_xor(sIn, m, 32);
        sTg += __shfl_xor(sTg, m, 32);
    }
    if ((t & 31) == 0) {
        const int slot = (b * 32 + chunk) * 4 + wave;
        ws[WS_PARTI + slot] = sIn;
        ws[WS_PARTT + slot] = sTg;
    }

    __syncthreads();
    // fold 128 private histograms -> per-WG histogram (fixed order => deterministic)
    if (t < 40) {
        const int img = t / 20, j = t % 20;
        float s = 0.0f;
        for (int u = 0; u < 128; ++u) s += histP[(img * 128 + u) * 20 + j];
        float* dst = ws + (img ? WS_HISTT : WS_HISTI);
        dst[(b * 32 + chunk) * 20 + j] = s;
    }
}

// ---------------- finalize: one wave32, WMMA row-sums for avg ----------------
typedef float v2f __attribute__((ext_vector_type(2)));
typedef float v8f __attribute__((ext_vector_type(8)));

__global__ void __launch_bounds__(32)
mlpa_finalize(float* __restrict__ out, const float* __restrict__ ws) {
    __shared__ float blkI[256], blkT[256];   // [b][blk] 16x16 block matrices
    __shared__ float avgI[16], avgT[16];

    const int lane = threadIdx.x;            // 0..31, EXEC all-ones (no divergence yet)

    // assemble block matrices from per-wave partials (8 chunks per block-row)
    #pragma unroll
    for (int e = lane; e < 256; e += 32) {
        const int b = e >> 4, idx = e & 15, br = idx >> 2, wv = idx & 3;
        float si = 0.0f, st = 0.0f;
        #pragma unroll
        for (int j = 0; j < 8; ++j) {
            const int slot = (b * 32 + (br * 8 + j)) * 4 + wv;
            si += ws[WS_PARTI + slot];
            st += ws[WS_PARTT + slot];
        }
        blkI[e] = si;
        blkT[e] = st;
    }
    __syncthreads();

    // avg[b] = sum_k blk[b][k] via 4 chained V_WMMA_F32_16X16X4_F32 with B = ones.
    // A layout (16x4 f32): lanes 0-15 hold (K,K+1), lanes 16-31 hold (K+2,K+3), M = lane&15.
    const int row  = lane & 15;
    const int koff = (lane >> 4) << 1;
    v8f accI = { 0.f, 0.f, 0.f, 0.f, 0.f, 0.f, 0.f, 0.f };
    v8f accT = { 0.f, 0.f, 0.f, 0.f, 0.f, 0.f, 0.f, 0.f };
    const v2f ones = { 1.0f, 1.0f };
    #pragma unroll
    for (int kc = 0; kc < 4; ++kc) {
        const int k = 4 * kc + koff;
        v2f aI; aI.x = blkI[row * 16 + k]; aI.y = blkI[row * 16 + k + 1];
        v2f aT; aT.x = blkT[row * 16 + k]; aT.y = blkT[row * 16 + k + 1];
        accI = __builtin_amdgcn_wmma_f32_16x16x4_f32(false, aI, false, ones,
                                                     (short)0, accI, false, false);
        accT = __builtin_amdgcn_wmma_f32_16x16x4_f32(false, aT, false, ones,
                                                     (short)0, accT, false, false);
    }
    // D[m][n] = avg[m] for every n. Pull column N=0 out via lanes 0 and 16.
    if (lane == 0 || lane == 16) {
        const int mb = (lane == 0) ? 0 : 8;
        #pragma unroll
        for (int j = 0; j < 8; ++j) { avgI[mb + j] = accI[j]; avgT[mb + j] = accT[j]; }
    }
    __syncthreads();

    const float hw = 262144.0f;
    float contrib = 0.0f, blkpart = 0.0f;
    if (lane < 16) {
        const int b = lane;
        // per-batch histogram (sum 32 chunk histograms, fixed order)
        float hs = 0.0f;
        for (int j = 0; j < 20; ++j) {
            float hi = 0.0f, ht = 0.0f;
            for (int c = 0; c < 32; ++c) {
                hi += ws[WS_HISTI + (b * 32 + c) * 20 + j];
                ht += ws[WS_HISTT + (b * 32 + c) * 20 + j];
            }
            const float d = hi / hw - ht / hw;
            hs += d * d;
        }
        const float mlpa_histo = hs / 16.0f;                    // / B
        const float ai = avgI[b], at = avgT[b];
        const float diff = ai - at;
        const float mlpa_avg = diff * diff / (hw * hw);
        const bool cond = (diff >= at * -0.4f) && (diff <= at * 0.4f);
        contrib = cond ? mlpa_histo : (mlpa_avg + mlpa_histo);

        const float s = 16.0f / hw;
        for (int k = 0; k < 16; ++k) {
            const float db = (blkI[b * 16 + k] - blkT[b * 16 + k]) * s;
            blkpart += db * db;
        }
    }
    float total = contrib + blkpart * (1.0f / 256.0f);          // mlpa_block = mean of 256
    #pragma unroll
    for (int m = 16; m >= 1; m >>= 1) total += __shfl_xor(total, m, 32);
    if (lane == 0) out[0] = total;
}

extern "C" void kernel_launch(void* const* d_in, const int* in_sizes, int n_in,
                              void* d_out, int out_size, void* d_ws, size_t ws_size,
                              hipStream_t stream) {
    const float* inp = (const float*)d_in[0];
    const float* tgt = (const float*)d_in[1];
    float* out = (float*)d_out;
    float* ws  = (float*)d_ws;   // needs 24576 floats (96 KB), fully rewritten each call

    mlpa_main<<<dim3(BATCH * 32), dim3(128), 0, stream>>>(inp, tgt, out, ws);
    mlpa_finalize<<<dim3(1), dim3(32), 0, stream>>>(out, ws);
}